// SELayer222_55379308314629
// MI455X (gfx1250) — compile-verified
//
#include <hip/hip_runtime.h>
#include <math.h>

// Problem constants (B,C,H,W) = (64,256,64,64), RED=16
#define BB   64
#define CCH  256
#define S_SP 4096          // H*W
#define HID  16            // C/RED

typedef float f4v __attribute__((ext_vector_type(4)));
typedef float v2f __attribute__((ext_vector_type(2)));
typedef float v8f __attribute__((ext_vector_type(8)));

// ---------------------------------------------------------------------------
// K1: z[b,s] = sum_c x[b,c,s] * conv_w[c]
// (1x1 conv commutes with the avg-pool, so do it BEFORE pooling: saves a full
//  pass over x). 256 blocks x 256 threads, each thread owns one float4 of s.
// Streams 256 MB; ~11 us at 23.3 TB/s.
// ---------------------------------------------------------------------------
__global__ void k_conv_z(const float* __restrict__ x, const float* __restrict__ w,
                         float* __restrict__ z) {
  int b  = blockIdx.x >> 2;                       // 4 blocks per batch
  int s0 = ((blockIdx.x & 3) << 10) + (threadIdx.x << 2);
  const float* xb = x + (size_t)b * CCH * S_SP + s0;
  float ax = 0.f, ay = 0.f, az = 0.f, aw = 0.f;
#pragma unroll 4
  for (int c = 0; c < CCH; ++c) {
    if (c + 16 < CCH)                              // uniform guard; emits global_prefetch_b8
      __builtin_prefetch(xb + (size_t)(c + 16) * S_SP, 0, 0);
    f4v v = *(const f4v*)(xb + (size_t)c * S_SP);  // global_load_b128, coalesced
    float wc = w[c];                               // uniform -> scalar load
    ax = fmaf(v.x, wc, ax);
    ay = fmaf(v.y, wc, ay);
    az = fmaf(v.z, wc, az);
    aw = fmaf(v.w, wc, aw);
  }
  f4v r; r.x = ax; r.y = ay; r.z = az; r.w = aw;
  *(f4v*)(z + (size_t)b * S_SP + s0) = r;
}

// ---------------------------------------------------------------------------
// K2: logits = avgpool3d(z) over (batch-in-group-of-8, h, w), /27 (+bias),
//     then softmax over the 4096 spatial positions. One block per batch.
// Tiny (3 MB total traffic).
// ---------------------------------------------------------------------------
__global__ void k_pool_softmax(const float* __restrict__ z, const float* __restrict__ bias,
                               float* __restrict__ attn) {
  __shared__ float zb[S_SP];    // z summed over the group-batch window
  __shared__ float red[256];
  int b = blockIdx.x, p = b & 7;
  int tid = threadIdx.x;
  const float* zp = z + (size_t)b * S_SP;
  for (int i = 0; i < 16; ++i) {
    int s = tid + (i << 8);
    float v = zp[s];
    if (p > 0) v += zp[s - S_SP];                  // zero pad outside [0,8)
    if (p < 7) v += zp[s + S_SP];
    zb[s] = v;
  }
  __syncthreads();

  const float inv27 = 1.0f / 27.0f;               // count_include_pad=True
  float bval = bias[0];
  float lg[16];
  float lmax = -3.0e38f;
  for (int i = 0; i < 16; ++i) {
    int s = tid + (i << 8);
    int h = s >> 6, w = s & 63;
    float sum = 0.f;
    for (int dh = -1; dh <= 1; ++dh) {
      int hh = h + dh;
      if ((unsigned)hh < 64u) {
        const float* row = &zb[hh << 6];
        int wlo = (w > 0) ? (w - 1) : 0;
        int whi = (w < 63) ? (w + 1) : 63;
        for (int ww = wlo; ww <= whi; ++ww) sum += row[ww];
      }
    }
    float l = fmaf(sum, inv27, bval);
    lg[i] = l;
    lmax = fmaxf(lmax, l);
  }
  // block max
  red[tid] = lmax; __syncthreads();
  for (int st = 128; st > 0; st >>= 1) {
    if (tid < st) red[tid] = fmaxf(red[tid], red[tid + st]);
    __syncthreads();
  }
  float gmax = red[0];
  __syncthreads();
  // exp + block sum
  float lsum = 0.f;
  for (int i = 0; i < 16; ++i) { lg[i] = expf(lg[i] - gmax); lsum += lg[i]; }
  red[tid] = lsum; __syncthreads();
  for (int st = 128; st > 0; st >>= 1) {
    if (tid < st) red[tid] += red[tid + st];
    __syncthreads();
  }
  float inv = 1.0f / red[0];
  float* ap = attn + (size_t)b * S_SP;
  for (int i = 0; i < 16; ++i) ap[tid + (i << 8)] = lg[i] * inv;
}

// ---------------------------------------------------------------------------
// K3: y[b,c] = sum_s x[b,c,s] * attn[b,s]. One wave per (b,c), 8 waves/block.
// attn staged in LDS (16 KB) once per block. Second 256 MB stream of x.
// ---------------------------------------------------------------------------
__global__ void k_wpool(const float* __restrict__ x, const float* __restrict__ attn,
                        float* __restrict__ y) {
  __shared__ f4v at4[S_SP / 4];
  int b = blockIdx.y;
  int tid = threadIdx.x;
  float* at = (float*)at4;
  const float* ap = attn + (size_t)b * S_SP;
  for (int i = 0; i < 16; ++i) at[tid + (i << 8)] = ap[tid + (i << 8)];
  __syncthreads();

  int wave = tid >> 5, lane = tid & 31;           // wave32
  int c = (blockIdx.x << 3) + wave;
  const f4v* xp = (const f4v*)(x + ((size_t)b * CCH + c) * S_SP);
  float acc = 0.f;
#pragma unroll 4
  for (int i = 0; i < 32; ++i) {
    f4v v = xp[(i << 5) + lane];                  // global_load_b128
    f4v a = at4[(i << 5) + lane];                 // ds_load_b128
    acc += v.x * a.x + v.y * a.y + v.z * a.z + v.w * a.w;
  }
  for (int off = 16; off > 0; off >>= 1) acc += __shfl_xor(acc, off, 32);
  if (lane == 0) y[b * CCH + c] = acc;
}

// ---------------------------------------------------------------------------
// K4: SE MLP with V_WMMA_F32_16X16X4_F32 (full fp32 precision).
//   hid  = relu(y @ fc1^T)     : (64x256)@(256x16), 4 M-tiles, K=256
//   gate = sigmoid(hid @ fc2^T): (64x16)@(16x256),  4 M-tiles x 16 N-tiles, K=16
// One block, 4 waves; wave i owns M-tile i.
// A layout (ISA 7.12.2, 32-bit A 16x4): lanes 0-15 -> M=lane, K = {0,1};
// lanes 16-31 -> M=lane-16, K = {2,3}.  D layout: VGPR v, lanes 0-15 -> M=v,
// lanes 16-31 -> M=v+8, N = lane&15.
// ---------------------------------------------------------------------------
__global__ void k_mlp_wmma(const float* __restrict__ y, const float* __restrict__ fc1,
                           const float* __restrict__ fc2, float* __restrict__ gate) {
  __shared__ float hid[BB][HID];
  int wave = threadIdx.x >> 5, lane = threadIdx.x & 31;
  int mrow = wave * 16 + (lane & 15);
  int kh   = (lane < 16) ? 0 : 2;
  int ncol = lane & 15;

  // GEMM1: hid = relu(y @ fc1^T); fc1 is (16,256) row-major -> B[k][n] = fc1[n*256+k]
  v8f c = {};
  for (int k0 = 0; k0 < CCH; k0 += 4) {
    v2f a, bm;
    a.x  = y[mrow * CCH + k0 + kh];
    a.y  = y[mrow * CCH + k0 + kh + 1];
    bm.x = fc1[ncol * CCH + k0 + kh];
    bm.y = fc1[ncol * CCH + k0 + kh + 1];
    c = __builtin_amdgcn_wmma_f32_16x16x4_f32(false, a, false, bm, (short)0, c, false, false);
  }
#pragma unroll
  for (int v = 0; v < 8; ++v) {
    int m = wave * 16 + v + ((lane < 16) ? 0 : 8);
    hid[m][ncol] = fmaxf(c[v], 0.f);
  }
  __syncthreads();

  // GEMM2: gate = sigmoid(hid @ fc2^T); fc2 is (256,16) row-major -> B[k][n] = fc2[n*16+k]
  for (int nt = 0; nt < 16; ++nt) {
    v8f d = {};
#pragma unroll
    for (int k0 = 0; k0 < HID; k0 += 4) {
      v2f a, bm;
      a.x  = hid[mrow][k0 + kh];
      a.y  = hid[mrow][k0 + kh + 1];
      int n2 = nt * 16 + ncol;
      bm.x = fc2[n2 * HID + k0 + kh];
      bm.y = fc2[n2 * HID + k0 + kh + 1];
      d = __builtin_amdgcn_wmma_f32_16x16x4_f32(false, a, false, bm, (short)0, d, false, false);
    }
#pragma unroll
    for (int v = 0; v < 8; ++v) {
      int m = wave * 16 + v + ((lane < 16) ? 0 : 8);
      gate[m * CCH + nt * 16 + ncol] = 1.0f / (1.0f + expf(-d[v]));
    }
  }
}

// ---------------------------------------------------------------------------
// K5: out = x * gate[b,c]. One block per (b,c) plane so the gate load is a
// scalar (uniform). Nontemporal on the 512 MB stream to spare L2.
// ---------------------------------------------------------------------------
__global__ void k_scale(const float* __restrict__ x, const float* __restrict__ gate,
                        float* __restrict__ out) {
  int bc = blockIdx.x;
  float g = gate[bc];                             // uniform -> s_load
  const f4v* xp = (const f4v*)(x + (size_t)bc * S_SP);
  f4v* op = (f4v*)(out + (size_t)bc * S_SP);
  int t = threadIdx.x;
#pragma unroll
  for (int i = 0; i < 4; ++i) {
    f4v v = __builtin_nontemporal_load(&xp[t + (i << 8)]);
    v.x *= g; v.y *= g; v.z *= g; v.w *= g;
    __builtin_nontemporal_store(v, &op[t + (i << 8)]);
  }
}

// ---------------------------------------------------------------------------
extern "C" void kernel_launch(void* const* d_in, const int* in_sizes, int n_in,
                              void* d_out, int out_size, void* d_ws, size_t ws_size,
                              hipStream_t stream) {
  const float* x     = (const float*)d_in[0];   // (64,256,64,64)
  const float* convw = (const float*)d_in[1];   // (256,)
  const float* convb = (const float*)d_in[2];   // (1,)
  const float* fc1   = (const float*)d_in[3];   // (16,256)
  const float* fc2   = (const float*)d_in[4];   // (256,16)
  float* out = (float*)d_out;

  float* ws   = (float*)d_ws;
  float* z    = ws;                  // B*S   = 262144 floats (1 MB)
  float* attn = ws + 262144;         // B*S   = 262144 floats (1 MB)
  float* yv   = ws + 524288;         // B*C   = 16384 floats
  float* gate = ws + 540672;         // B*C   = 16384 floats

  k_conv_z      <<<BB * 4,            256, 0, stream>>>(x, convw, z);
  k_pool_softmax<<<BB,                256, 0, stream>>>(z, convb, attn);
  k_wpool       <<<dim3(CCH / 8, BB), 256, 0, stream>>>(x, attn, yv);
  k_mlp_wmma    <<<1,                 128, 0, stream>>>(yv, fc1, fc2, gate);
  k_scale       <<<BB * CCH,          256, 0, stream>>>(x, gate, out);
}